// ChebNet_71339406786681
// MI455X (gfx1250) — compile-verified
//
#include <hip/hip_runtime.h>
#include <cstdint>
#include <cstddef>

// ---------------- problem constants (match reference) ----------------
#define NNODES 50000
#define NEDGES 800000
#define INC    128
#define HIDC   128
#define OUTC   64
#define KORD   4

typedef __attribute__((ext_vector_type(2))) float v2f;
typedef __attribute__((ext_vector_type(8))) float v8f;

// Guaranteed-native fp32 scatter atomic: GLOBAL_ATOMIC_ADD_F32 (no return ->
// STOREcnt path), device scope so results are L2-coherent across the grid.
__device__ __forceinline__ void atomAddF(float* p, float v) {
  asm volatile("global_atomic_add_f32 %0, %1, off scope:SCOPE_DEV"
               :: "v"(p), "v"(v)
               : "memory");
}

// ---------------- elementwise helpers ----------------
__global__ void k_zero(float* __restrict__ p, int n) {
  int i = blockIdx.x * blockDim.x + threadIdx.x;
  if (i < n) p[i] = 0.0f;
}
__global__ void k_copy(float* __restrict__ dst, const float* __restrict__ src, int n) {
  int i = blockIdx.x * blockDim.x + threadIdx.x;
  if (i < n) dst[i] = src[i];
}
__global__ void k_copyneg(float* __restrict__ dst, const float* __restrict__ src, int n) {
  int i = blockIdx.x * blockDim.x + threadIdx.x;
  if (i < n) dst[i] = -src[i];
}

// ---------------- graph preprocessing ----------------
// in-degree by destination (float counts, atomics into L2)
__global__ void k_deg(const long long* __restrict__ ei, float* __restrict__ deg) {
  int e = blockIdx.x * blockDim.x + threadIdx.x;
  if (e >= NEDGES) return;
  int d = (int)ei[NEDGES + e];
  atomAddF(&deg[d], 1.0f);
}

// w[e] = -rsqrt(deg[src]) * rsqrt(deg[dst]), 0 where deg==0
__global__ void k_wedge(const long long* __restrict__ ei, const float* __restrict__ deg,
                        float* __restrict__ w) {
  int e = blockIdx.x * blockDim.x + threadIdx.x;
  if (e >= NEDGES) return;
  int s = (int)ei[e];
  int d = (int)ei[NEDGES + e];
  float ds = deg[s], dd = deg[d];
  float is = (ds > 0.0f) ? rsqrtf(fmaxf(ds, 1.0f)) : 0.0f;
  float id = (dd > 0.0f) ? rsqrtf(fmaxf(dd, 1.0f)) : 0.0f;
  w[e] = -is * id;
}

// out[dst] += scale * w[e] * h[src]  over 128 features; 32 threads/edge, float4 per thread.
// Gather hits the 192MB L2 (25.6MB working set); scatter is native fp32 L2 atomics.
__global__ void k_prop(const long long* __restrict__ ei, const float* __restrict__ w,
                       const float* __restrict__ h, float* __restrict__ out, float scale) {
  int tid = blockIdx.x * blockDim.x + threadIdx.x;
  int e = tid >> 5;
  int q = tid & 31;
  if (e >= NEDGES) return;
  int s = (int)ei[e];
  int d = (int)ei[NEDGES + e];
  float wv = scale * w[e];
  const float4 hv = *(const float4*)(h + (size_t)s * INC + q * 4);
  float* op = out + (size_t)d * INC + q * 4;
  atomAddF(op + 0, wv * hv.x);
  atomAddF(op + 1, wv * hv.y);
  atomAddF(op + 2, wv * hv.z);
  atomAddF(op + 3, wv * hv.w);
}

// ---------------- Chebyshev contraction: out = act( sum_k Tx[k] @ W[k] + b ) ----------------
// A: [4][NNODES][Cin] (Tx stack), Wt: [4][Cin][Cout], bias: [Cout], out: [NNODES][Cout]
// Block = 128 threads = 4 wave32s; each wave computes one 16x16 tile with
// v_wmma_f32_16x16x4_f32, reducing over 4*Cin. W column-block staged in LDS.
__global__ void __launch_bounds__(128)
k_cheb_gemm(const float* __restrict__ A, const float* __restrict__ Wt,
            const float* __restrict__ bias, float* __restrict__ out,
            int Cin, int Cout, int do_relu) {
  __shared__ float sW[KORD * 128 * 16];  // 32 KB: [4*Cin][16] column block

  const int colbase = blockIdx.y * 16;
  const int kd_total = KORD * Cin;  // 512
  for (int t = threadIdx.x; t < kd_total * 16; t += blockDim.x) {
    int r = t >> 4;       // k*Cin + kin
    int c = t & 15;
    sW[t] = Wt[(size_t)r * Cout + colbase + c];
  }
  __syncthreads();

  const int wave = threadIdx.x >> 5;
  const int lane = threadIdx.x & 31;
  const int half = lane >> 4;    // selects K pair {0,1} vs {2,3}
  const int l15  = lane & 15;

  const int mtile = blockIdx.x * 4 + wave;
  if (mtile * 16 >= NNODES) return;  // wave-uniform guard: EXEC all-ones inside WMMA

  const int arow = mtile * 16 + l15;  // A: lane holds M = lane&15 (both halves)
  v8f acc = {};

  for (int k = 0; k < KORD; ++k) {
    const float* Ak = A + ((size_t)k * NNODES + arow) * Cin;
    const float* Ws = sW + k * Cin * 16;
    for (int kk = 0; kk < Cin; kk += 4) {
      // A 16x4 f32 layout: VGPR0/1 = K {2*half, 2*half+1}
      v2f a = *(const v2f*)(Ak + kk + 2 * half);
      // B 4x16 f32 layout: VGPR0/1 = K {2*half, 2*half+1}, N = lane&15
      v2f b;
      b.x = Ws[(kk + 2 * half) * 16 + l15];
      b.y = Ws[(kk + 2 * half + 1) * 16 + l15];
      acc = __builtin_amdgcn_wmma_f32_16x16x4_f32(false, a, false, b,
                                                  (short)0, acc, false, false);
    }
  }

  // C/D layout: VGPR r -> M = r + 8*half, N = lane&15
  float bv = bias[colbase + l15];
  for (int r = 0; r < 8; ++r) {
    int orow = mtile * 16 + half * 8 + r;
    float v = acc[r] + bv;
    if (do_relu) v = fmaxf(v, 0.0f);
    out[(size_t)orow * Cout + colbase + l15] = v;
  }
}

// ---------------- row-wise log_softmax over 64 logits (one wave per node) ----------------
__global__ void k_logsoftmax(const float* __restrict__ in, float* __restrict__ out) {
  int gid = blockIdx.x * blockDim.x + threadIdx.x;
  int node = gid >> 5;
  int lane = gid & 31;
  if (node >= NNODES) return;
  float v0 = in[(size_t)node * OUTC + lane];
  float v1 = in[(size_t)node * OUTC + lane + 32];
  float m = fmaxf(v0, v1);
  for (int off = 16; off > 0; off >>= 1) m = fmaxf(m, __shfl_xor(m, off, 32));
  float s = expf(v0 - m) + expf(v1 - m);
  for (int off = 16; off > 0; off >>= 1) s += __shfl_xor(s, off, 32);
  float ls = logf(s);
  out[(size_t)node * OUTC + lane]      = v0 - m - ls;
  out[(size_t)node * OUTC + lane + 32] = v1 - m - ls;
}

// ---------------- driver ----------------
static inline int NB(long long n, int b) { return (int)((n + b - 1) / b); }

extern "C" void kernel_launch(void* const* d_in, const int* in_sizes, int n_in,
                              void* d_out, int out_size, void* d_ws, size_t ws_size,
                              hipStream_t stream) {
  (void)in_sizes; (void)n_in; (void)out_size; (void)ws_size;
  const float*     x  = (const float*)d_in[0];
  const long long* ei = (const long long*)d_in[1];  // [2][NEDGES] int64
  const float*     W1 = (const float*)d_in[2];      // [4][128][128]
  const float*     b1 = (const float*)d_in[3];      // [128]
  const float*     W2 = (const float*)d_in[4];      // [4][128][64]
  const float*     b2 = (const float*)d_in[5];      // [64]
  float* out = (float*)d_out;                       // [NNODES][64]

  // workspace layout (floats)
  float* ws  = (float*)d_ws;
  float* deg = ws;                                   // NNODES
  float* we  = deg + NNODES;                         // NEDGES
  float* tx  = we + NEDGES;                          // 4 * NNODES * 128
  const size_t NC = (size_t)NNODES * 128;
  float* tx0 = tx;
  float* tx1 = tx + NC;
  float* tx2 = tx + 2 * NC;
  float* tx3 = tx + 3 * NC;
  float* h1  = tx + 4 * NC;                          // NNODES * 128
  float* lg  = h1 + NC;                              // NNODES * 64

  const int ncInt = (int)NC;
  const int propBlocks = NB((long long)NEDGES * 32, 256);

  // ---- edge normalization ----
  k_zero<<<NB(NNODES, 256), 256, 0, stream>>>(deg, NNODES);
  k_deg<<<NB(NEDGES, 256), 256, 0, stream>>>(ei, deg);
  k_wedge<<<NB(NEDGES, 256), 256, 0, stream>>>(ei, deg, we);

  // ---- layer 1: build Tx stack (Tx2 = 2*prop(Tx1) - Tx0 folded into init+scaled scatter) ----
  k_copy<<<NB(ncInt, 256), 256, 0, stream>>>(tx0, x, ncInt);
  k_zero<<<NB(ncInt, 256), 256, 0, stream>>>(tx1, ncInt);
  k_prop<<<propBlocks, 256, 0, stream>>>(ei, we, tx0, tx1, 1.0f);
  k_copyneg<<<NB(ncInt, 256), 256, 0, stream>>>(tx2, tx0, ncInt);
  k_prop<<<propBlocks, 256, 0, stream>>>(ei, we, tx1, tx2, 2.0f);
  k_copyneg<<<NB(ncInt, 256), 256, 0, stream>>>(tx3, tx1, ncInt);
  k_prop<<<propBlocks, 256, 0, stream>>>(ei, we, tx2, tx3, 2.0f);

  {
    dim3 grid(NB(NNODES / 16, 4), HIDC / 16);
    k_cheb_gemm<<<grid, 128, 0, stream>>>(tx, W1, b1, h1, INC, HIDC, 1);
  }

  // ---- layer 2 ----
  k_copy<<<NB(ncInt, 256), 256, 0, stream>>>(tx0, h1, ncInt);
  k_zero<<<NB(ncInt, 256), 256, 0, stream>>>(tx1, ncInt);
  k_prop<<<propBlocks, 256, 0, stream>>>(ei, we, tx0, tx1, 1.0f);
  k_copyneg<<<NB(ncInt, 256), 256, 0, stream>>>(tx2, tx0, ncInt);
  k_prop<<<propBlocks, 256, 0, stream>>>(ei, we, tx1, tx2, 2.0f);
  k_copyneg<<<NB(ncInt, 256), 256, 0, stream>>>(tx3, tx1, ncInt);
  k_prop<<<propBlocks, 256, 0, stream>>>(ei, we, tx2, tx3, 2.0f);

  {
    dim3 grid(NB(NNODES / 16, 4), OUTC / 16);
    k_cheb_gemm<<<grid, 128, 0, stream>>>(tx, W2, b2, lg, HIDC, OUTC, 0);
  }

  // ---- log_softmax ----
  k_logsoftmax<<<NB((long long)NNODES * 32, 256), 256, 0, stream>>>(lg, out);
}